// PCEN_25082609009148
// MI455X (gfx1250) — compile-verified
//
#include <hip/hip_runtime.h>
#include <math.h>

// PCEN: per-channel energy normalization with a parallel (chunked affine) scan
// over the time axis. Bandwidth-bound: ~205 MB moved, floor ~8.8us @ 23.3 TB/s.
// No matmul -> no WMMA; CDNA5 value = wave32 shuffles, 320KB LDS, async LDS DMA.

#define AS1 __attribute__((address_space(1)))
#define AS3 __attribute__((address_space(3)))

typedef int v4i_t __attribute__((ext_vector_type(4)));

#if defined(__gfx1250__) && __has_builtin(__builtin_amdgcn_global_load_async_to_lds_b128) && __has_builtin(__builtin_amdgcn_s_wait_asynccnt)
#define PCEN_ASYNC_LOAD 1
#endif
#if defined(__gfx1250__) && __has_builtin(__builtin_amdgcn_global_store_async_from_lds_b128) && __has_builtin(__builtin_amdgcn_s_wait_asynccnt)
#define PCEN_ASYNC_STORE 1
#endif

constexpr int kT     = 10000;   // time steps per row
constexpr int kM     = 80;      // mel channels
constexpr int kBlock = 256;     // 8 wave32 per block
constexpr int kChunk = 40;      // 256*40 = 10240 >= 10000
constexpr int kVec4  = kT / 4;  // 2500 16-byte packets per row (40000B, 16B aligned)
constexpr float kEps = 1e-6f;

__global__ __launch_bounds__(kBlock) void pcen_scan_kernel(
    const float* __restrict__ mel,
    const float* __restrict__ log_s,
    const float* __restrict__ log_alpha,
    const float* __restrict__ log_delta,
    const float* __restrict__ log_r,
    float* __restrict__ out)
{
    __shared__ __align__(16) float row[kBlock * kChunk];   // 40 KB staging buffer
    __shared__ float wp[kBlock / 32];
    __shared__ float wc[kBlock / 32];

    const int rowid = blockIdx.x;          // 0 .. B*M-1
    const int m     = rowid % kM;          // channel (uniform per block -> SGPR loads)
    const int tid   = threadIdx.x;

    // Per-channel parameters (sigmoid/exp + clip), uniform across the block.
    const float s  = fminf(fmaxf(1.f / (1.f + __expf(-log_s[m])),     0.05f), 0.3f);
    const float al = fminf(fmaxf(1.f / (1.f + __expf(-log_alpha[m])), 0.9f),  0.999f);
    const float de = fminf(fmaxf(__expf(log_delta[m]),                0.001f), 0.1f);
    const float rr = fminf(fmaxf(1.f / (1.f + __expf(-log_r[m])),     0.05f), 0.25f);
    const float a  = 1.f - s;

    const float* g = mel + (size_t)rowid * kT;

    // ---- Stage the whole row into LDS (coalesced 16B/lane, async DMA) ----
#if PCEN_ASYNC_LOAD
    for (int i = tid; i < kVec4; i += kBlock) {
        __builtin_amdgcn_global_load_async_to_lds_b128(
            (AS1 v4i_t*)(g + i * 4), (AS3 v4i_t*)(&row[i * 4]), 0, 0);
    }
    __builtin_amdgcn_s_wait_asynccnt(0);
#else
    for (int i = tid; i < kVec4; i += kBlock) {
        *(float4*)(&row[i * 4]) = *(const float4*)(g + i * 4);
    }
#endif
    __syncthreads();

    // ---- Pass 1: reduce each 40-element chunk to an affine map x -> P*x + C ----
    const int begin = tid * kChunk;
    int end = begin + kChunk; if (end > kT) end = kT;   // threads 250..255 are identity

    float p = 1.f, c = 0.f;
    for (int j = begin; j < end; ++j) {
        c = __fmaf_rn(a, c, s * row[j]);
        p *= a;
    }

    // ---- Inclusive affine scan: wave32 shuffles, then 8-entry cross-wave combine ----
    float P = p, C = c;
    const int lane = tid & 31;
#pragma unroll
    for (int off = 1; off < 32; off <<= 1) {
        float Pp = __shfl_up(P, (unsigned)off);
        float Cp = __shfl_up(C, (unsigned)off);
        if (lane >= off) {
            C = __fmaf_rn(P, Cp, C);   // compose: later ∘ earlier
            P *= Pp;
        }
    }
    const int w = tid >> 5;
    if (lane == 31) { wp[w] = P; wc[w] = C; }
    __syncthreads();

    float Pw = 1.f, Cw = 0.f;                 // exclusive prefix over earlier waves
    for (int i = 0; i < w; ++i) {
        Cw = __fmaf_rn(wp[i], Cw, wc[i]);
        Pw *= wp[i];
    }
    float Pe = __shfl_up(P, 1u);              // exclusive within wave
    float Ce = __shfl_up(C, 1u);
    if (lane == 0) { Pe = 1.f; Ce = 0.f; }

    // Carry into this chunk; scan seed is mel[...,0] (= row[0]).
    const float m0 = row[0];
    float sm = __fmaf_rn(Pe, __fmaf_rn(Pw, m0, Cw), Ce);

    // ---- Pass 2: replay chunk with true carry, apply PCEN, overwrite LDS ----
    const float na = -al;
    const float dr = __powf(de, rr);
    for (int j = begin; j < end; ++j) {
        float x   = row[j];
        sm        = __fmaf_rn(a, sm, s * x);
        float gn  = __powf(kEps + sm, na);             // v_log_f32 + v_exp_f32
        float y   = __powf(__fmaf_rn(x, gn, de), rr) - dr;
        row[j] = y;
    }
    __syncthreads();

    // ---- Stream results back out, coalesced 16B/lane (async DMA) ----
    float* o = out + (size_t)rowid * kT;
#if PCEN_ASYNC_STORE
    for (int i = tid; i < kVec4; i += kBlock) {
        __builtin_amdgcn_global_store_async_from_lds_b128(
            (AS1 v4i_t*)(o + i * 4), (AS3 v4i_t*)(&row[i * 4]), 0, 0);
    }
    __builtin_amdgcn_s_wait_asynccnt(0);
#else
    for (int i = tid; i < kVec4; i += kBlock) {
        *(float4*)(o + i * 4) = *(const float4*)(&row[i * 4]);
    }
#endif
}

extern "C" void kernel_launch(void* const* d_in, const int* in_sizes, int n_in,
                              void* d_out, int out_size, void* d_ws, size_t ws_size,
                              hipStream_t stream)
{
    const float* mel       = (const float*)d_in[0];
    const float* log_s     = (const float*)d_in[1];
    const float* log_alpha = (const float*)d_in[2];
    const float* log_delta = (const float*)d_in[3];
    const float* log_r     = (const float*)d_in[4];
    float* out             = (float*)d_out;

    const int rows = in_sizes[0] / kT;   // B*M = 2560 rows, one block each
    pcen_scan_kernel<<<rows, kBlock, 0, stream>>>(
        mel, log_s, log_alpha, log_delta, log_r, out);
    (void)d_ws; (void)ws_size; (void)n_in; (void)out_size;
}